// GaussianLoss_12524124635800
// MI455X (gfx1250) — compile-verified
//
#include <hip/hip_runtime.h>
#include <hip/hip_bf16.h>

typedef float v2f __attribute__((ext_vector_type(2)));
typedef float v8f __attribute__((ext_vector_type(8)));

#define WMMA_F32_4(A, B, C) \
    __builtin_amdgcn_wmma_f32_16x16x4_f32(false, (A), false, (B), (short)0, (C), false, false)

constexpr int IMG_W = 3840;
constexpr int IMG_H = 2160;
constexpr int IMG_C = 3;
constexpr int TX = IMG_W / 16;                   // 240
constexpr int TY = IMG_H / 16;                   // 135
constexpr int TILES_PER_C = TX * TY;             // 32400
constexpr int TOTAL_TILES = TILES_PER_C * IMG_C; // 97200
constexpr int WAVES_PER_BLOCK = 8;
constexpr int TAB = 64;                          // zero-padded tap table (indices t+15, t in [-15,48])
// per-wave LDS floats: in_a 32x32 + in_b 32x32 + mid 5x(32x16)
constexpr int PERW = 1024 + 1024 + 5 * 512;      // 4608 floats = 18 KB
constexpr float C1V = 0.0001f;                   // 0.01^2
constexpr float C2V = 0.0009f;                   // 0.03^2

// Normalized 11-tap Gaussian, sigma = 1.5 (symmetric). Returns G[t] for t in [0,10], else 0.
// Only used once per block to build the LDS tap table.
__device__ __forceinline__ float gtap(int t) {
    float r = 0.0f;
    r = (t == 0 || t == 10) ? 0.00102838f : r;
    r = (t == 1 || t == 9)  ? 0.00759888f : r;
    r = (t == 2 || t == 8)  ? 0.03600077f : r;
    r = (t == 3 || t == 7)  ? 0.10936060f : r;
    r = (t == 4 || t == 6)  ? 0.21300553f : r;
    r = (t == 5)            ? 0.26601190f : r;
    return r;
}

__global__ void zero_accum_kernel(float* ws) {
    ws[0] = 0.0f;   // sum of ssim_map
    ws[1] = 0.0f;   // sum of |a-b|
}

__global__ __launch_bounds__(256)
void ssim_l1_kernel(const float* __restrict__ pred,
                    const float* __restrict__ targ,
                    float* __restrict__ ws) {
    extern __shared__ float smem[];

    const int lane = threadIdx.x & 31;
    const int wid  = threadIdx.x >> 5;
    const int n16  = lane & 15;
    const int hi   = (lane >> 4) & 1;    // 0 for lanes 0-15, 1 for lanes 16-31
    const int hi2  = hi * 2;
    const int hi8  = hi * 8;

    // ---- build zero-padded Gaussian tap table in LDS (shared by all 8 waves)
    float* tab = smem;                   // TAB floats; tab[i] = g[i-15] (0 outside [0,10])
    if (threadIdx.x < TAB) {
        tab[threadIdx.x] = gtap((int)threadIdx.x - 15);
    }
    __syncthreads();

    long gw = (long)blockIdx.x * WAVES_PER_BLOCK + wid;
    if (gw >= TOTAL_TILES) return;       // wave-uniform (grid is exact anyway)

    const int c   = (int)(gw / TILES_PER_C);
    const int rem = (int)(gw % TILES_PER_C);
    const int ty  = rem / TX;
    const int tx  = rem % TX;
    const int x0  = tx * 16;
    const int y0  = ty * 16;

    const float* pa = pred + (size_t)c * (size_t)IMG_H * (size_t)IMG_W;
    const float* pb = targ + (size_t)c * (size_t)IMG_H * (size_t)IMG_W;

    float* ina = smem + TAB + wid * PERW;   // 32 rows x 32 cols (halo region, zero-padded)
    float* inb = ina + 1024;
    float* mid = inb + 1024;                // 5 maps x 32 rows x 16 cols

    // ---- load 26x28 halo region (rows y0-5..y0+20, cols x0-5..x0+22), zero padded to 32x32
    for (int r = 0; r < 32; ++r) {
        const int gy = y0 + r - 5;
        const int gx = x0 + lane - 5;
        float va = 0.0f, vb = 0.0f;
        if (r < 26 && lane < 28 &&
            (unsigned)gy < (unsigned)IMG_H && (unsigned)gx < (unsigned)IMG_W) {
            const size_t o = (size_t)gy * IMG_W + gx;
            va = pa[o];
            vb = pb[o];
        }
        ina[r * 32 + lane] = va;
        inb[r * 32 + lane] = vb;
    }

    // ---- per-lane banded Gaussian weight fragments, gathered from the LDS table
    // Wh[k][c] = g[k-c+5] (horizontal pass, B operand), Wv[m][r] = g[r-m] (vertical pass, A operand)
    float whv0[7], whv1[7], wvv0[7], wvv1[7];
#pragma unroll
    for (int q = 0; q < 7; ++q) {
        const int tw = 4 * q + hi2 - n16 + 15;   // in [0, 42]
        wvv0[q] = tab[tw];
        wvv1[q] = tab[tw + 1];
        whv0[q] = tab[tw + 5];
        whv1[q] = tab[tw + 6];
    }

    // ---- stage 1: horizontal pass, Mid = In * Wh, two 16-row blocks, K=28 in 7 chunks
#pragma unroll
    for (int R = 0; R < 2; ++R) {
        v8f ca = {}, cb = {}, caa = {}, cbb = {}, cab = {};
#pragma unroll
        for (int q = 0; q < 7; ++q) {
            const int off = (R * 16 + n16) * 32 + 4 * q + hi2;   // 8B aligned
            v2f a2 = *reinterpret_cast<const v2f*>(ina + off);
            v2f b2 = *reinterpret_cast<const v2f*>(inb + off);
            v2f B;  B.x = whv0[q];  B.y = whv1[q];
            v2f Aaa = a2 * a2;
            v2f Abb = b2 * b2;
            v2f Aab = a2 * b2;
            ca  = WMMA_F32_4(a2,  B, ca);
            cb  = WMMA_F32_4(b2,  B, cb);
            caa = WMMA_F32_4(Aaa, B, caa);
            cbb = WMMA_F32_4(Abb, B, cbb);
            cab = WMMA_F32_4(Aab, B, cab);
        }
        // D layout: VGPR v holds row v (lanes 0-15) / row v+8 (lanes 16-31), col = lane&15
#pragma unroll
        for (int v = 0; v < 8; ++v) {
            const int mrow = R * 16 + hi8 + v;
            const int o = mrow * 16 + n16;
            mid[0 * 512 + o] = ca[v];
            mid[1 * 512 + o] = cb[v];
            mid[2 * 512 + o] = caa[v];
            mid[3 * 512 + o] = cbb[v];
            mid[4 * 512 + o] = cab[v];
        }
    }

    // ---- stage 2: vertical pass, Out = Wv * Mid, K=28 in 7 chunks
    v8f outm[5];
#pragma unroll
    for (int m = 0; m < 5; ++m) {
        const float* md = mid + m * 512;
        v8f acc = {};
#pragma unroll
        for (int q = 0; q < 7; ++q) {
            const int k0 = 4 * q + hi2;
            v2f A;  A.x = wvv0[q];  A.y = wvv1[q];
            v2f B;  B.x = md[k0 * 16 + n16];
                    B.y = md[(k0 + 1) * 16 + n16];
            acc = WMMA_F32_4(A, B, acc);
        }
        outm[m] = acc;
    }

    // ---- per-pixel SSIM + L1 (C/D fragments of all 5 maps are lane/element aligned)
    float ssim_s = 0.0f, l1_s = 0.0f;
#pragma unroll
    for (int i = 0; i < 8; ++i) {
        const float mu1 = outm[0][i];
        const float mu2 = outm[1][i];
        const float m11 = mu1 * mu1;
        const float m22 = mu2 * mu2;
        const float m12 = mu1 * mu2;
        const float s1  = outm[2][i] - m11;
        const float s2  = outm[3][i] - m22;
        const float s12 = outm[4][i] - m12;
        const float num = (2.0f * m12 + C1V) * (2.0f * s12 + C2V);
        const float den = (m11 + m22 + C1V) * (s1 + s2 + C2V);
        ssim_s += num * __builtin_amdgcn_rcpf(den);   // v_rcp_f32 instead of full IEEE divide

        const int p  = lane + 32 * i;        // 256 tile pixels spread over 32 lanes
        const int pr = p >> 4;
        const int pc = p & 15;
        const int o  = (pr + 5) * 32 + (pc + 5);
        l1_s += fabsf(ina[o] - inb[o]);
    }

    // ---- wave32 reduction, then one atomic per wave
#pragma unroll
    for (int off = 16; off >= 1; off >>= 1) {
        ssim_s += __shfl_xor(ssim_s, off, 32);
        l1_s   += __shfl_xor(l1_s,   off, 32);
    }
    if (lane == 0) {
        unsafeAtomicAdd(&ws[0], ssim_s);
        unsafeAtomicAdd(&ws[1], l1_s);
    }
}

__global__ void finalize_kernel(const float* __restrict__ ws, float* __restrict__ out) {
    const float invN = 1.0f / (float)(IMG_C * IMG_H * IMG_W);   // 1/24883200
    const float ssim_mean = ws[0] * invN;
    const float l1_mean   = ws[1] * invN;
    out[0] = 0.8f * l1_mean + 0.2f * (1.0f - ssim_mean);
}

extern "C" void kernel_launch(void* const* d_in, const int* in_sizes, int n_in,
                              void* d_out, int out_size, void* d_ws, size_t ws_size,
                              hipStream_t stream) {
    (void)in_sizes; (void)n_in; (void)out_size; (void)ws_size;
    const float* pred = (const float*)d_in[0];
    const float* targ = (const float*)d_in[1];
    float* ws  = (float*)d_ws;
    float* out = (float*)d_out;

    zero_accum_kernel<<<1, 1, 0, stream>>>(ws);

    const int blocks = (TOTAL_TILES + WAVES_PER_BLOCK - 1) / WAVES_PER_BLOCK;      // 12150
    const size_t shmem = (size_t)(TAB + WAVES_PER_BLOCK * PERW) * sizeof(float);   // 147712 B
    ssim_l1_kernel<<<blocks, 256, shmem, stream>>>(pred, targ, ws);

    finalize_kernel<<<1, 1, 0, stream>>>(ws, out);
}